// GroupContrastLoss_57389353009479
// MI455X (gfx1250) — compile-verified
//
#include <hip/hip_runtime.h>
#include <hip/hip_bf16.h>
#include <cstdint>

// ---------------- problem constants ----------------
#define BS    16
#define C_    64
#define H_    128
#define W_    128
#define HW    (H_ * W_)          // 16384
#define NC    80
#define TOPK  128
#define KTOT  (TOPK + NC)        // 208
#define TAU   0.07f
#define EPSN  1e-12f

typedef float v2f __attribute__((ext_vector_type(2)));
typedef float v8f __attribute__((ext_vector_type(8)));

// ---------------- workspace layout (bytes) ----------------
// feat_n     : BS*C*HW f32
// q          : BS*NC*C f32
// k0n        : BS*NC*C f32
// class_mask : BS*NC f32
// max_values : BS*HW f32
// class_idx  : BS*HW i32
// topk_val   : BS*TOPK f32
// topk_idx   : BS*TOPK i32
// topk_class : BS*TOPK i32
// topk_mask  : BS*TOPK f32
// k_all      : BS*KTOT*C f32
// loss_bn    : BS*NC f32
#define OFF_FEATN   ((size_t)0)
#define OFF_Q       (OFF_FEATN + (size_t)BS*C_*HW*4)
#define OFF_K0      (OFF_Q     + (size_t)BS*NC*C_*4)
#define OFF_CM      (OFF_K0    + (size_t)BS*NC*C_*4)
#define OFF_MAXV    (OFF_CM    + (size_t)BS*NC*4)
#define OFF_CIDX    (OFF_MAXV  + (size_t)BS*HW*4)
#define OFF_TKV     (OFF_CIDX  + (size_t)BS*HW*4)
#define OFF_TKI     (OFF_TKV   + (size_t)BS*TOPK*4)
#define OFF_TKC     (OFF_TKI   + (size_t)BS*TOPK*4)
#define OFF_TKM     (OFF_TKC   + (size_t)BS*TOPK*4)
#define OFF_KALL    (OFF_TKM   + (size_t)BS*TOPK*4)
#define OFF_LBN     (OFF_KALL  + (size_t)BS*KTOT*C_*4)

// ============================================================
// K1: L2-normalize feat along channel dim (per spatial location)
// ============================================================
__global__ void k_normalize_feat(const float* __restrict__ feat,
                                 float* __restrict__ feat_n) {
    int idx = blockIdx.x * blockDim.x + threadIdx.x;   // over BS*HW
    if (idx >= BS * HW) return;
    int b = idx / HW, p = idx % HW;
    const float* src = feat   + (size_t)b * C_ * HW + p;
    float*       dst = feat_n + (size_t)b * C_ * HW + p;
    float v[C_];
    float ss = 0.f;
#pragma unroll
    for (int c = 0; c < C_; ++c) { v[c] = src[(size_t)c * HW]; ss += v[c] * v[c]; }
    float inv = 1.0f / fmaxf(sqrtf(ss), EPSN);
#pragma unroll
    for (int c = 0; c < C_; ++c) dst[(size_t)c * HW] = v[c] * inv;
}

// ============================================================
// K2: per-(b,class) hm peak scan -> q, k0 (normalized), class_mask
// one 256-thread block per (b,n)
// ============================================================
__global__ void k_peaks(const float* __restrict__ hm,
                        const float* __restrict__ feat_n,
                        float* __restrict__ q,
                        float* __restrict__ k0n,
                        float* __restrict__ class_mask) {
    __shared__ float s_val[256];
    __shared__ int   s_pos[256];
    __shared__ float s_k[C_];
    __shared__ float s_inv;

    int bn = blockIdx.x;           // b*NC + n
    int b  = bn / NC;
    int tid = threadIdx.x;
    const float* hmp = hm + (size_t)bn * HW;

    float best = 0.f; int bpos = -1;
    for (int p = tid; p < HW; p += 256) {
        float v = hmp[p];
        if (v > best) { best = v; bpos = p; }
    }
    s_val[tid] = best; s_pos[tid] = bpos;
    __syncthreads();
    for (int s = 128; s > 0; s >>= 1) {
        if (tid < s) {
            float vo = s_val[tid + s]; int po = s_pos[tid + s];
            if (vo > s_val[tid] || (vo == s_val[tid] && po >= 0 &&
                                    (s_pos[tid] < 0 || po < s_pos[tid]))) {
                s_val[tid] = vo; s_pos[tid] = po;
            }
        }
        __syncthreads();
    }
    int   pos = s_pos[0];
    float hv  = s_val[0];

    // gather feat at peak, build raw k0 = hv * feat
    if (tid < C_) {
        float kv = 0.f;
        if (pos >= 0) kv = hv * feat_n[((size_t)b * C_ + tid) * HW + pos];
        s_k[tid] = kv;
    }
    __syncthreads();
    if (tid == 0) {
        float ss = 0.f;
        for (int c = 0; c < C_; ++c) ss += s_k[c] * s_k[c];
        s_inv = 1.0f / fmaxf(sqrtf(ss), EPSN);
        class_mask[bn] = (pos >= 0 && hv == 1.0f) ? 1.0f : 0.0f;
    }
    __syncthreads();
    if (tid < C_) {
        float fv = (pos >= 0) ? feat_n[((size_t)b * C_ + tid) * HW + pos] : 0.f;
        q[(size_t)bn * C_ + tid]   = (pos >= 0 && hv == 1.0f) ? fv : 0.f;
        k0n[(size_t)bn * C_ + tid] = s_k[tid] * s_inv;
    }
}

// ============================================================
// K3: fused hm_free + masked score max/argmax per pixel
// ============================================================
__global__ void k_score_max(const float* __restrict__ score,
                            const float* __restrict__ hm,
                            const float* __restrict__ class_mask,
                            float* __restrict__ max_values,
                            int*   __restrict__ class_idx) {
    __shared__ float cm[NC];
    int idx = blockIdx.x * blockDim.x + threadIdx.x;   // over BS*HW
    int b = idx / HW, p = idx % HW;
    if (threadIdx.x < NC) cm[threadIdx.x] = class_mask[b * NC + threadIdx.x];
    __syncthreads();
    float hmsum = 0.f, best = 0.f; int bestn = 0;
    for (int n = 0; n < NC; ++n) {
        size_t off = ((size_t)b * NC + n) * HW + p;
        hmsum += hm[off];
        float s = score[off] * cm[n];
        if (s > best) { best = s; bestn = n; }
    }
    if (hmsum != 0.f) { best = 0.f; bestn = 0; }
    max_values[idx] = best;
    class_idx[idx]  = bestn;
}

// ============================================================
// K4: per-batch iterative top-128 in LDS (lowest-index tie-break)
// dyn LDS: HW floats + 256 floats + 256 ints
// ============================================================
__global__ void k_topk(const float* __restrict__ max_values,
                       const int*   __restrict__ class_idx,
                       float* __restrict__ topk_val,
                       int*   __restrict__ topk_idx,
                       int*   __restrict__ topk_class,
                       float* __restrict__ topk_mask) {
    extern __shared__ float sm[];
    float* vals = sm;                 // HW
    float* rv   = sm + HW;            // 256
    int*   ri   = (int*)(sm + HW + 256);

    int b = blockIdx.x, tid = threadIdx.x;
    for (int p = tid; p < HW; p += 256) vals[p] = max_values[(size_t)b * HW + p];
    __syncthreads();

    for (int t = 0; t < TOPK; ++t) {
        float best = -1.f; int bpos = HW;
        for (int p = tid; p < HW; p += 256) {
            float v = vals[p];
            if (v > best) { best = v; bpos = p; }   // strided => first hit is lowest idx
        }
        rv[tid] = best; ri[tid] = bpos;
        __syncthreads();
        for (int s = 128; s > 0; s >>= 1) {
            if (tid < s) {
                float vo = rv[tid + s]; int po = ri[tid + s];
                if (vo > rv[tid] || (vo == rv[tid] && po < ri[tid])) {
                    rv[tid] = vo; ri[tid] = po;
                }
            }
            __syncthreads();
        }
        if (tid == 0) {
            int pos = ri[0]; float val = rv[0];
            topk_val[b * TOPK + t]   = val;
            topk_idx[b * TOPK + t]   = pos;
            topk_class[b * TOPK + t] = class_idx[(size_t)b * HW + pos];
            topk_mask[b * TOPK + t]  = (val > 0.f) ? 1.0f : 0.0f;
            vals[pos] = -1.0f;
        }
        __syncthreads();
    }
}

// ============================================================
// K5: build k_all = [k_top (gathered feat) ; k0n]   [BS][KTOT][C]
// one 32-lane row-copier per row, 8 rows per block
// ============================================================
__global__ void k_build_kall(const float* __restrict__ feat_n,
                             const float* __restrict__ k0n,
                             const int*   __restrict__ topk_idx,
                             float* __restrict__ k_all) {
    int tid  = threadIdx.x;
    int row  = blockIdx.x * 8 + (tid >> 5);
    int lane = tid & 31;
    if (row >= BS * KTOT) return;
    int b = row / KTOT, k = row % KTOT;
    float* dst = k_all + ((size_t)b * KTOT + k) * C_;
    if (k < TOPK) {
        int pos = topk_idx[b * TOPK + k];
        for (int c = lane; c < C_; c += 32)
            dst[c] = feat_n[((size_t)b * C_ + c) * HW + pos];
    } else {
        int n = k - TOPK;
        for (int c = lane; c < C_; c += 32)
            dst[c] = k0n[((size_t)b * NC + n) * C_ + c];
    }
}

// ============================================================
// K6a: zero entire output buffer; K6b: scatter pseudo_hm
// ============================================================
__global__ void k_zero(float* __restrict__ out, size_t n) {
    size_t i = (size_t)blockIdx.x * blockDim.x + threadIdx.x;
    if (i < n) out[i] = 0.0f;
}

__global__ void k_scatter_pseudo(const int* __restrict__ topk_idx,
                                 const int* __restrict__ topk_class,
                                 const float* __restrict__ topk_mask,
                                 float* __restrict__ pseudo) {
    int i = blockIdx.x * blockDim.x + threadIdx.x;   // BS*TOPK
    if (i >= BS * TOPK) return;
    int b = i / TOPK;
    if (topk_mask[i] > 0.f) {
        int cls = topk_class[i], pos = topk_idx[i];
        pseudo[((size_t)b * NC + cls) * HW + pos] = 0.9f;
    }
}

// ============================================================
// K7: per-batch contrastive: WMMA f32 16x16x4 GEMM (80x208x64)
//     into LDS, then fused masked exp/log-sum row reduction.
// grid = BS, block = 256 (8 waves), dyn LDS = NC*KTOT floats
// ============================================================
__global__ void k_contrastive(const float* __restrict__ q,
                              const float* __restrict__ k_all,
                              const float* __restrict__ class_mask,
                              const int*   __restrict__ topk_class,
                              const float* __restrict__ topk_mask,
                              float* __restrict__ loss_bn) {
    extern __shared__ float sim[];               // [NC][KTOT]
    int b    = blockIdx.x;
    int tid  = threadIdx.x;
    int wave = tid >> 5;
    int lane = tid & 31;

    const float* qb = q     + (size_t)b * NC * C_;
    const float* kb = k_all + (size_t)b * KTOT * C_;

    // ---- phase A: sim[m][n] = q[m,:] . k_all[n,:] via V_WMMA_F32_16X16X4_F32
    const int MT = NC / 16;        // 5
    const int NT = KTOT / 16;      // 13
    int lrow  = lane & 15;         // M (A) / N (B) index within tile
    int khalf = (lane >> 4) * 2;   // K sub-offset per ISA A/B layout

    for (int tile = wave; tile < MT * NT; tile += 8) {   // uniform per wave
        int mt = tile / NT, nt = tile % NT;
        const float* arow = qb + ((size_t)(mt * 16 + lrow)) * C_;
        const float* brow = kb + ((size_t)(nt * 16 + lrow)) * C_;
        v8f acc = {};
        for (int kk = 0; kk < C_; kk += 4) {
            v2f a, bb;
            a.x  = arow[kk + khalf];
            a.y  = arow[kk + khalf + 1];
            bb.x = brow[kk + khalf];
            bb.y = brow[kk + khalf + 1];
            acc = __builtin_amdgcn_wmma_f32_16x16x4_f32(
                false, a, false, bb, (short)0, acc, false, false);
        }
        // C/D layout: VGPR r -> M = r + 8*(lane>>4), N = lane&15
        int mbase = mt * 16 + (lane >> 4) * 8;
        int ncol  = nt * 16 + (lane & 15);
#pragma unroll
        for (int r = 0; r < 8; ++r)
            sim[(mbase + r) * KTOT + ncol] = acc[r];
    }
    __syncthreads();

    // ---- phase B: per class-row masked softmax-style log loss
    const float invtau = 1.0f / TAU;
    for (int n = wave; n < NC; n += 8) {
        float cmn = class_mask[b * NC + n];
        float sum_e = 0.f, sum_s = 0.f; int cnt = 0;
        for (int k = lane; k < KTOT; k += 32) {
            float s = sim[n * KTOT + k] * invtau;
            float kam, km;
            if (k < TOPK) {
                kam = topk_mask[b * TOPK + k];
                km  = (kam > 0.f && topk_class[b * TOPK + k] == n && cmn > 0.f)
                      ? 1.f : 0.f;
            } else {
                kam = class_mask[b * NC + (k - TOPK)];
                km  = ((k - TOPK) == n && cmn > 0.f) ? 1.f : 0.f;
            }
            sum_e += expf(s) * kam;
            if (km > 0.f) { cnt += 1; sum_s += s; }
        }
        for (int off = 16; off > 0; off >>= 1) {
            sum_e += __shfl_xor(sum_e, off, 32);
            sum_s += __shfl_xor(sum_s, off, 32);
            cnt   += __shfl_xor(cnt,   off, 32);
        }
        if (lane == 0) {
            float ssum = (sum_e == 0.f) ? 1.0f : sum_e;
            float ll   = sum_s - (float)cnt * logf(ssum);   // sum of log(sim/sum)
            float num  = (cnt == 0) ? 1.0f : (float)cnt;
            loss_bn[b * NC + n] = (ll / num) * cmn;
        }
    }
}

// ============================================================
// K8: deterministic final scalar reduce -> d_out[0] = -loss
// ============================================================
__global__ void k_final(const float* __restrict__ loss_bn,
                        const float* __restrict__ class_mask,
                        float* __restrict__ out_loss) {
    __shared__ float sl[256], sc[256];
    int tid = threadIdx.x;
    float l = 0.f, c = 0.f;
    for (int i = tid; i < BS * NC; i += 256) { l += loss_bn[i]; c += class_mask[i]; }
    sl[tid] = l; sc[tid] = c;
    __syncthreads();
    for (int s = 128; s > 0; s >>= 1) {
        if (tid < s) { sl[tid] += sl[tid + s]; sc[tid] += sc[tid + s]; }
        __syncthreads();
    }
    if (tid == 0) {
        float cnt = (sc[0] == 0.f) ? 1.0f : sc[0];
        out_loss[0] = -(sl[0] / cnt);
    }
}

// ============================================================
extern "C" void kernel_launch(void* const* d_in, const int* in_sizes, int n_in,
                              void* d_out, int out_size, void* d_ws, size_t ws_size,
                              hipStream_t stream) {
    (void)in_sizes; (void)n_in; (void)out_size; (void)ws_size;
    const float* feat  = (const float*)d_in[0];
    const float* score = (const float*)d_in[1];
    const float* hm    = (const float*)d_in[2];

    char* ws = (char*)d_ws;
    float* feat_n     = (float*)(ws + OFF_FEATN);
    float* q          = (float*)(ws + OFF_Q);
    float* k0n        = (float*)(ws + OFF_K0);
    float* class_mask = (float*)(ws + OFF_CM);
    float* max_values = (float*)(ws + OFF_MAXV);
    int*   class_idx  = (int*)  (ws + OFF_CIDX);
    float* topk_val   = (float*)(ws + OFF_TKV);
    int*   topk_idx   = (int*)  (ws + OFF_TKI);
    int*   topk_class = (int*)  (ws + OFF_TKC);
    float* topk_mask  = (float*)(ws + OFF_TKM);
    float* k_all      = (float*)(ws + OFF_KALL);
    float* loss_bn    = (float*)(ws + OFF_LBN);

    float* out   = (float*)d_out;      // [0] = -loss, [1..] = pseudo_hm
    float* pseudo = out + 1;

    // K1 normalize feat
    k_normalize_feat<<<(BS * HW + 255) / 256, 256, 0, stream>>>(feat, feat_n);

    // K2 peaks -> q, k0n, class_mask
    k_peaks<<<BS * NC, 256, 0, stream>>>(hm, feat_n, q, k0n, class_mask);

    // K3 masked score max/argmax per pixel (fused hm_free)
    k_score_max<<<(BS * HW + 255) / 256, 256, 0, stream>>>(
        score, hm, class_mask, max_values, class_idx);

    // K4 per-batch top-128
    size_t topk_lds = (size_t)HW * 4 + 256 * 4 + 256 * 4;
    k_topk<<<BS, 256, topk_lds, stream>>>(max_values, class_idx,
                                          topk_val, topk_idx, topk_class, topk_mask);

    // K5 k_all = [k_top ; k0]
    k_build_kall<<<(BS * KTOT + 7) / 8, 256, 0, stream>>>(feat_n, k0n, topk_idx, k_all);

    // K6 zero output, scatter pseudo heatmap
    size_t out_n = (size_t)1 + (size_t)BS * NC * HW;
    k_zero<<<(unsigned)((out_n + 255) / 256), 256, 0, stream>>>(out, out_n);
    k_scatter_pseudo<<<(BS * TOPK + 255) / 256, 256, 0, stream>>>(
        topk_idx, topk_class, topk_mask, pseudo);

    // K7 WMMA contrastive per batch
    size_t sim_lds = (size_t)NC * KTOT * 4;
    k_contrastive<<<BS, 256, sim_lds, stream>>>(q, k_all, class_mask,
                                                topk_class, topk_mask, loss_bn);

    // K8 final scalar
    k_final<<<1, 256, 0, stream>>>(loss_bn, class_mask, out);
}